// StateMatrixEncoder_111669150333
// MI455X (gfx1250) — compile-verified
//
#include <hip/hip_runtime.h>
#include <hip/hip_bf16.h>

// B=8, L=1024, K=5, S=256, H=512
// out: [8,1024,5,512] f32; ws: pooled [8192,512] f32 (16 MB)

typedef float v2f __attribute__((ext_vector_type(2)));
typedef float v8f __attribute__((ext_vector_type(8)));

#define LCONV 1024
#define HDIM  512

// ---------------------------------------------------------------------------
// Kernel 1: fused gather (sm_tail -> out channels 1..4) + hold-prev + mean
// pool -> pooled (workspace). One block per (b,l); 256 threads x float2 = 512.
// ---------------------------------------------------------------------------
__global__ __launch_bounds__(256) void sme_prep(
    const float* __restrict__ session,   // [8,4,256,512]
    const int*   __restrict__ stm,       // [8,1024,5]
    float*       __restrict__ out,       // [8,1024,5,512]
    float*       __restrict__ pooled)    // [8192,512]
{
    const int m   = blockIdx.x;          // b*1024 + l
    const int b   = m >> 10;
    const int l   = m & 1023;
    const int h0  = threadIdx.x * 2;

    float sx = 0.f, sy = 0.f;
#pragma unroll
    for (int j = 0; j < 4; ++j) {
        const int sc = stm[m * 5 + 1 + j];
        const int sp = (l > 0) ? stm[(m - 1) * 5 + 1 + j] : -1;

        float2 smv = make_float2(0.f, 0.f);   // sm_tail value (0 when held)
        float2 orv = make_float2(0.f, 0.f);   // one_res value
        if (sc > 0) {
            const float* src =
                session + (((size_t)(b * 4 + j)) * 256 + (size_t)(sc - 1)) * HDIM + h0;
            smv = *(const float2*)src;
            orv = smv;
        } else if (sp > 0) {
            // hold: one_res = sm_tail[l-1] = gathered at l-1 (0 if l-1 also held)
            const float* src =
                session + (((size_t)(b * 4 + j)) * 256 + (size_t)(sp - 1)) * HDIM + h0;
            orv = *(const float2*)src;
        }
        *(float2*)(out + ((size_t)m * 5 + 1 + j) * HDIM + h0) = smv;
        sx += orv.x;
        sy += orv.y;
    }
    *(float2*)(pooled + (size_t)m * HDIM + h0) = make_float2(sx * 0.25f, sy * 0.25f);
}

// ---------------------------------------------------------------------------
// Kernel 2: row0 = relu(concat(pooled, conv_state) @ W + bias) -> out ch 0.
// X is [8192,1024]: k<512 -> pooled[m,k]; k>=512 -> conv[b, max(l-1,0), k-512].
// Block tile: M=64, N=128; 8 waves, each wave: 16-wide N strip, 4 M sub-tiles.
// f32 WMMA 16x16x4. A tile staged in LDS (stride 20 floats: 16B-aligned b128
// stores, conflict-free b64 fragment reads).
// ---------------------------------------------------------------------------
__global__ __launch_bounds__(256) void sme_gemm(
    const float* __restrict__ pooled,    // [8192,512]
    const float* __restrict__ conv,      // [8,1024,512]
    const float* __restrict__ Wm,        // [1024,512]
    const float* __restrict__ bias,      // [512]
    float*       __restrict__ out)       // [8192,5,512] (channel 0)
{
    __shared__ float As[64 * 20];

    const int tid    = threadIdx.x;
    const int wave   = tid >> 5;
    const int lane   = tid & 31;
    const int half   = lane >> 4;        // 0: K{0,1} / M rows 0..7 ; 1: K{2,3} / M rows 8..15
    const int lane16 = lane & 15;

    const int m0   = blockIdx.x * 64;
    const int ncol = blockIdx.y * 128 + wave * 16 + lane16;

    v8f acc[4] = {};

    const int ldr = tid >> 2;            // 0..63 : row within M tile
    const int ldc = (tid & 3) << 2;      // 0,4,8,12 : col within K stage

    for (int k0 = 0; k0 < 1024; k0 += 16) {
        // ---- cooperative A-tile load: X[m0..m0+63, k0..k0+15] ----
        const int gc   = k0 + ldc;
        const int grow = m0 + ldr;
        float4 v;
        if (gc < HDIM) {
            v = *(const float4*)(pooled + (size_t)grow * HDIM + gc);
        } else {
            const int ll = grow & (LCONV - 1);
            const int bb = grow >> 10;
            const int sl = (ll == 0) ? 0 : (ll - 1);
            v = *(const float4*)(conv + ((size_t)(bb << 10) + sl) * HDIM + (gc - HDIM));
        }
        __syncthreads();                               // prior stage consumed
        *(float4*)(&As[ldr * 20 + ldc]) = v;           // 16B-aligned (20*4=80 bytes/row)
        __syncthreads();

        // ---- 4 WMMA k-steps over this stage ----
#pragma unroll
        for (int kk = 0; kk < 4; ++kk) {
            const int kb = k0 + kk * 4 + half * 2;
            v2f bf;                                    // B[k][n]: elem v = K(half*2+v), N=lane16
            bf.x = Wm[(size_t)kb * HDIM + ncol];
            bf.y = Wm[(size_t)(kb + 1) * HDIM + ncol];
#pragma unroll
            for (int ms = 0; ms < 4; ++ms) {
                // A frag: lane = half*16+m holds A[m][half*2 + {0,1}]
                v2f af = *(const v2f*)(&As[(ms * 16 + lane16) * 20 + kk * 4 + half * 2]);
                acc[ms] = __builtin_amdgcn_wmma_f32_16x16x4_f32(
                    false, af, false, bf, (short)0, acc[ms], false, false);
            }
        }
    }

    // ---- epilogue: bias + relu, scatter to out[..., 0, :] ----
    const float bv = bias[ncol];
#pragma unroll
    for (int ms = 0; ms < 4; ++ms) {
#pragma unroll
        for (int r = 0; r < 8; ++r) {
            const int mrow = m0 + ms * 16 + half * 8 + r;   // C layout: vgpr r, half -> M
            float val = acc[ms][r] + bv;
            val = fmaxf(val, 0.f);
            out[(size_t)mrow * (5 * HDIM) + ncol] = val;
        }
    }
}

// ---------------------------------------------------------------------------
extern "C" void kernel_launch(void* const* d_in, const int* in_sizes, int n_in,
                              void* d_out, int out_size, void* d_ws, size_t ws_size,
                              hipStream_t stream) {
    (void)in_sizes; (void)n_in; (void)out_size; (void)ws_size;
    // setup_inputs order:
    // 0: utterance_repre (unused by reference)
    const float* conv    = (const float*)d_in[1];   // [8,1024,512]
    const float* session = (const float*)d_in[2];   // [8,4,256,512]
    const int*   stm     = (const int*)d_in[3];     // [8,1024,5]
    // 4: max_conversation_length (scalar, unused)
    const float* Wm      = (const float*)d_in[5];   // [1024,512]
    const float* bias    = (const float*)d_in[6];   // [512]

    float* out    = (float*)d_out;
    float* pooled = (float*)d_ws;                   // 8192*512*4 = 16 MB scratch

    sme_prep<<<8192, 256, 0, stream>>>(session, stm, out, pooled);

    dim3 grid(8192 / 64, 512 / 128);                // (128, 4)
    sme_gemm<<<grid, 256, 0, stream>>>(pooled, conv, Wm, bias, out);
}